// PointerNetwork_73684458930772
// MI455X (gfx1250) — compile-verified
//
#include <hip/hip_runtime.h>
#include <math.h>

typedef float v2f __attribute__((ext_vector_type(2)));
typedef float v8f __attribute__((ext_vector_type(8)));

#define B_ 16
#define T_ 512
#define S_ 4096
#define D_ 256
#define NEG_INF_ (-1.0e9f)

// Native CDNA5 V_TANH_F32 if the toolchain exposes it; OCML fallback otherwise.
#if __has_builtin(__builtin_amdgcn_tanhf)
#define FAST_TANH(x) __builtin_amdgcn_tanhf(x)
#elif __has_builtin(__builtin_amdgcn_tanh_f32)
#define FAST_TANH(x) __builtin_amdgcn_tanh_f32(x)
#else
#define FAST_TANH(x) tanhf(x)
#endif

// ---------------------------------------------------------------------------
// C[M,256] = A[M,256] @ W[256,256] + bias[256] (fp32, V_WMMA_F32_16X16X4_F32).
// 128 threads = 4 waves. Each wave owns a 64(M) x 16(N) strip: 4 independent
// 16x16 accumulators sharing one B-fragment per K-step. K and N are the
// compile-time constant 256 so all fragment addresses are ONE base VGPR pair
// plus immediate offsets -> global_load (not flat), clause-able, LOADcnt only.
// A-frag (16x4 MxK): lanes 0-15 rows M=l, K={k,k+1}; lanes 16-31 K={k+2,k+3}.
// B-frag (4x16 KxN): VGPR0 row K=k(+2), VGPR1 row K=k+1(+3), N = lane&15.
// C tile: VGPR r -> row r (lanes 0-15) / row r+8 (lanes 16-31), N = lane&15.
// ---------------------------------------------------------------------------
__global__ __launch_bounds__(128)
void wmma_gemm_f32(const float* __restrict__ A, const float* __restrict__ W,
                   const float* __restrict__ bias, float* __restrict__ C,
                   int M) {
  const int lane = threadIdx.x & 31;
  const int wave = threadIdx.x >> 5;
  const int n0 = blockIdx.x * 16;
  const int m0 = blockIdx.y * 256 + wave * 64;
  const int half = lane >> 4;   // 0 -> K pair {0,1}; 1 -> K pair {2,3}
  const int l    = lane & 15;

  v8f acc0 = {}, acc1 = {}, acc2 = {}, acc3 = {};

  // Single base pointers; all other addressing via immediate offsets.
  const float* __restrict__ ab = A + (size_t)(m0 + l) * D_ + half * 2;
  const float* __restrict__ wb = W + (size_t)(half * 2) * D_ + n0 + l;

#pragma unroll 8
  for (int k = 0; k < D_; k += 4) {
    // Load all five fragments into distinct registers first so the loads form
    // a clause and the WMMAs ride staggered s_wait_loadcnt values.
    v2f b, a0, a1, a2, a3;
    b[0]  = wb[k * D_];
    b[1]  = wb[k * D_ + D_];
    a0[0] = ab[k];
    a0[1] = ab[k + 1];
    a1[0] = ab[k + 16 * D_];
    a1[1] = ab[k + 16 * D_ + 1];
    a2[0] = ab[k + 32 * D_];
    a2[1] = ab[k + 32 * D_ + 1];
    a3[0] = ab[k + 48 * D_];
    a3[1] = ab[k + 48 * D_ + 1];

    acc0 = __builtin_amdgcn_wmma_f32_16x16x4_f32(false, a0, false, b,
                                                 (short)0, acc0, false, false);
    acc1 = __builtin_amdgcn_wmma_f32_16x16x4_f32(false, a1, false, b,
                                                 (short)0, acc1, false, false);
    acc2 = __builtin_amdgcn_wmma_f32_16x16x4_f32(false, a2, false, b,
                                                 (short)0, acc2, false, false);
    acc3 = __builtin_amdgcn_wmma_f32_16x16x4_f32(false, a3, false, b,
                                                 (short)0, acc3, false, false);
  }

  const float bn = bias[n0 + l];
  float* __restrict__ cb = C + (size_t)(m0 + half * 8) * D_ + n0 + l;
#pragma unroll
  for (int r = 0; r < 8; ++r) {
    cb[(size_t)(r) * D_]      = acc0[r] + bn;
    cb[(size_t)(r + 16) * D_] = acc1[r] + bn;
    cb[(size_t)(r + 32) * D_] = acc2[r] + bn;
    cb[(size_t)(r + 48) * D_] = acc3[r] + bn;
  }
}

// ---------------------------------------------------------------------------
// Scores: e[b,t,s] = sum_d tanh(qp[b,t,d] + k_proj[b,s,d]) * v[d]
// Mask-independent -> all T steps in parallel. Tile 16 t x 32 s per 256-thread
// block; qp/k tiles + v staged in LDS (k row stride padded to 257 floats:
// 257 % 64 == 1 -> lanes hit distinct banks). Each thread produces two (t,s)
// scores, reusing each k element and v element across both t rows.
// ---------------------------------------------------------------------------
__global__ __launch_bounds__(256)
void score_kernel(const float* __restrict__ qp,     // [B*T, D]
                  const float* __restrict__ kproj,  // [B*S, D]
                  const float* __restrict__ v,      // [D]
                  float* __restrict__ e_out) {      // [B, T, S]
  __shared__ float qp_s[16][D_];
  __shared__ float k_s[32 * (D_ + 1)];
  __shared__ float v_s[D_];

  const int tid = threadIdx.x;
  const int s0 = blockIdx.x * 32;
  const int t0 = blockIdx.y * 16;
  const int b  = blockIdx.z;

  for (int i = tid; i < 16 * D_; i += 256) {
    const int r = i >> 8, c = i & (D_ - 1);
    qp_s[r][c] = qp[(size_t)(b * T_ + t0 + r) * D_ + c];
  }
  for (int i = tid; i < 32 * D_; i += 256) {
    const int r = i >> 8, c = i & (D_ - 1);
    k_s[r * (D_ + 1) + c] = kproj[(size_t)(b * S_ + s0 + r) * D_ + c];
  }
  v_s[tid] = v[tid];
  __syncthreads();

  const int tt = tid >> 5;   // 0..7 (also handles tt+8)
  const int ss = tid & 31;   // 0..31
  float acc0 = 0.0f, acc1 = 0.0f;
  const float* krow = &k_s[ss * (D_ + 1)];
#pragma unroll 4
  for (int d = 0; d < D_; ++d) {
    const float kv = krow[d];
    const float vv = v_s[d];
    acc0 += FAST_TANH(qp_s[tt][d] + kv) * vv;
    acc1 += FAST_TANH(qp_s[tt + 8][d] + kv) * vv;
  }
  e_out[(size_t)(b * T_ + t0 + tt) * S_ + s0 + ss] = acc0;
  e_out[(size_t)(b * T_ + t0 + tt + 8) * S_ + s0 + ss] = acc1;
}

// ---------------------------------------------------------------------------
// Sequential greedy pointer loop: one block per batch, 512 steps.
// In-place: e[b,t,:] (already in d_out) -> masked log_softmax; emits argmax.
// ---------------------------------------------------------------------------
__global__ __launch_bounds__(256)
void step_kernel(float* __restrict__ logp,      // [B, T, S], holds e on entry
                 float* __restrict__ idx_out) { // [B, T] (indices as floats)
  __shared__ unsigned int mask[S_ / 32];  // 128 words
  __shared__ float red[256];
  __shared__ int  redi[256];

  const int b = blockIdx.x;
  const int tid = threadIdx.x;
  const int PER = S_ / 256;  // 16 elements per thread

  for (int i = tid; i < S_ / 32; i += 256) mask[i] = 0u;
  __syncthreads();

  float* base = logp + (size_t)b * T_ * S_;
  for (int t = 0; t < T_; ++t) {
    float* e = base + (size_t)t * S_;

    float vals[16];
    float lmax = -INFINITY;
    int   lidx = 0;
    for (int j = 0; j < PER; ++j) {
      const int s = tid * PER + j;
      float x = e[s];
      const bool m = (mask[s >> 5] >> (s & 31)) & 1u;
      x = m ? NEG_INF_ : x;
      vals[j] = x;
      if (x > lmax) { lmax = x; lidx = s; }  // first-max within thread
    }
    red[tid] = lmax;
    redi[tid] = lidx;
    __syncthreads();
    for (int off = 128; off > 0; off >>= 1) {
      if (tid < off) {
        const float o = red[tid + off];
        const int  oi = redi[tid + off];
        if (o > red[tid] || (o == red[tid] && oi < redi[tid])) {
          red[tid] = o;
          redi[tid] = oi;
        }
      }
      __syncthreads();
    }
    const float gmax = red[0];
    const int   gidx = redi[0];
    __syncthreads();

    float lsum = 0.0f;
    for (int j = 0; j < PER; ++j) lsum += expf(vals[j] - gmax);
    red[tid] = lsum;
    __syncthreads();
    for (int off = 128; off > 0; off >>= 1) {
      if (tid < off) red[tid] += red[tid + off];
      __syncthreads();
    }
    const float logZ = gmax + logf(red[0]);
    __syncthreads();

    for (int j = 0; j < PER; ++j) e[tid * PER + j] = vals[j] - logZ;

    if (tid == 0) {
      idx_out[(size_t)b * T_ + t] = (float)gidx;
      mask[gidx >> 5] |= (1u << (gidx & 31));
    }
    __syncthreads();
  }
}

// ---------------------------------------------------------------------------
extern "C" void kernel_launch(void* const* d_in, const int* in_sizes, int n_in,
                              void* d_out, int out_size, void* d_ws, size_t ws_size,
                              hipStream_t stream) {
  const float* dec = (const float*)d_in[0];  // [B,T,D]
  const float* enc = (const float*)d_in[1];  // [B,S,D]
  const float* Wq  = (const float*)d_in[2];  // [D,D]
  const float* bq  = (const float*)d_in[3];  // [D]
  const float* Wk  = (const float*)d_in[4];  // [D,D]
  const float* bk  = (const float*)d_in[5];  // [D]
  const float* v   = (const float*)d_in[6];  // [D]

  float* out_logp = (float*)d_out;                         // [B,T,S]
  float* out_idx  = out_logp + (size_t)B_ * T_ * S_;       // [B,T]

  float* qp    = (float*)d_ws;                             // [B*T, D]  8 MB
  float* kproj = qp + (size_t)B_ * T_ * D_;                // [B*S, D] 67 MB

  // 1) qp = dec @ Wq + bq   (M = B*T = 8192)
  {
    dim3 grid(D_ / 16, (B_ * T_) / 256);
    wmma_gemm_f32<<<grid, 128, 0, stream>>>(dec, Wq, bq, qp, B_ * T_);
  }
  // 2) kproj = enc @ Wk + bk (M = B*S = 65536)
  {
    dim3 grid(D_ / 16, (B_ * S_) / 256);
    wmma_gemm_f32<<<grid, 128, 0, stream>>>(enc, Wk, bk, kproj, B_ * S_);
  }
  // 3) all scores in parallel -> write straight into logp region of d_out
  {
    dim3 grid(S_ / 32, T_ / 16, B_);
    score_kernel<<<grid, 256, 0, stream>>>(qp, kproj, v, out_logp);
  }
  // 4) sequential mask / log_softmax / argmax per batch
  step_kernel<<<B_, 256, 0, stream>>>(out_logp, out_idx);
}